// UWE_22514218566139
// MI455X (gfx1250) — compile-verified
//
#include <hip/hip_runtime.h>
#include <hip/hip_bf16.h>
#include <float.h>

#define T_DIM   20
#define K_DIM   50
#define V_DIM   50000
#define D_DIM   200
#define NTOP    20
#define M_DIM   (K_DIM * NTOP)          // 1000 candidates per time slice
#define N_TILES ((M_DIM + 15) / 16)     // 63
#define INV_TEMP (1.0f / 0.07f)
#define NEG_BIG  (-1e30f)

typedef __attribute__((ext_vector_type(2))) float v2f;
typedef __attribute__((ext_vector_type(8))) float v8f;

// Only check the builtin in the DEVICE pass; the host pass (x86) never has it.
#if defined(__HIP_DEVICE_COMPILE__)
#if !__has_builtin(__builtin_amdgcn_wmma_f32_16x16x4_f32)
#error "wmma_f32_16x16x4_f32 builtin not available for this offload target"
#endif
#endif

// ---------------------------------------------------------------------------
// Kernel A: per (t,k) row, top-20 indices of beta[t,k,:] over V=50000.
// One block per row; per-thread register top-20 + LDS merge (20 argmax rounds).
// ---------------------------------------------------------------------------
__global__ __launch_bounds__(256)
void uwe_topk(const float* __restrict__ beta, int* __restrict__ idxOut) {
    const int row = blockIdx.x;                 // t*K + k
    const float* b = beta + (size_t)row * V_DIM;
    const int tid = threadIdx.x;

    float tv[NTOP]; int ti[NTOP];
#pragma unroll
    for (int j = 0; j < NTOP; ++j) { tv[j] = -FLT_MAX; ti[j] = 0x7fffffff; }

    for (int i = tid; i < V_DIM; i += 256) {
        float v = b[i];
        if (v > tv[NTOP - 1]) {                 // rare after warm-up
            float cv = v; int ci = i;
#pragma unroll
            for (int p = 0; p < NTOP; ++p) {    // bubble-insert, all static idx
                if (cv > tv[p]) {
                    float fv = tv[p]; int fi = ti[p];
                    tv[p] = cv; ti[p] = ci; cv = fv; ci = fi;
                }
            }
        }
    }

    __shared__ float sv[256 * NTOP];
    __shared__ int   si[256 * NTOP];
    __shared__ float rv[256];
    __shared__ int   rslot[256];
    __shared__ int   rw[256];

#pragma unroll
    for (int j = 0; j < NTOP; ++j) { sv[tid * NTOP + j] = tv[j]; si[tid * NTOP + j] = ti[j]; }
    __syncthreads();

    for (int r = 0; r < NTOP; ++r) {
        float bv = -FLT_MAX; int bs = tid * NTOP; int bw = 0x7fffffff;
        for (int j = 0; j < NTOP; ++j) {
            int slot = tid * NTOP + j;
            float v = sv[slot]; int w = si[slot];
            if (v > bv || (v == bv && w < bw)) { bv = v; bw = w; bs = slot; }
        }
        rv[tid] = bv; rslot[tid] = bs; rw[tid] = bw;
        __syncthreads();
        for (int s = 128; s > 0; s >>= 1) {
            if (tid < s) {
                float ov = rv[tid + s]; int ow = rw[tid + s];
                if (ov > rv[tid] || (ov == rv[tid] && ow < rw[tid])) {
                    rv[tid] = ov; rw[tid] = ow; rslot[tid] = rslot[tid + s];
                }
            }
            __syncthreads();
        }
        if (tid == 0) { idxOut[row * NTOP + r] = rw[0]; sv[rslot[0]] = -FLT_MAX; }
        __syncthreads();
    }
}

// ---------------------------------------------------------------------------
// Kernel B: validity per (t, m): wordcount==0 and first occurrence in slice.
// ---------------------------------------------------------------------------
__global__ __launch_bounds__(256)
void uwe_valid(const int* __restrict__ wc, const int* __restrict__ idx,
               unsigned char* __restrict__ valid, int* __restrict__ flags) {
    const int t = blockIdx.x;
    __shared__ int sIdx[M_DIM];
    __shared__ int sAny;
    if (threadIdx.x == 0) sAny = 0;
    for (int m = threadIdx.x; m < M_DIM; m += 256) sIdx[m] = idx[t * M_DIM + m];
    __syncthreads();
    int any = 0;
    for (int m = threadIdx.x; m < M_DIM; m += 256) {
        int w = sIdx[m];
        int ok = (wc[(size_t)t * V_DIM + w] == 0);
        if (ok) {
            for (int j = 0; j < m; ++j) if (sIdx[j] == w) { ok = 0; break; }
        }
        valid[t * M_DIM + m] = (unsigned char)ok;
        any |= ok;
    }
    if (any) atomicOr(&sAny, 1);
    __syncthreads();
    if (threadIdx.x == 0) flags[t] = sAny;
}

// ---------------------------------------------------------------------------
// Kernel C: per t, sim = (topic_emb[t] x gathered word_emb^T)/TEMP via
// v_wmma_f32_16x16x4_f32, masked online logsumexp over 1000 candidates,
// mean over 50 topics.  Block = 512 threads = 16 waves, one block per t.
// 32-bit A/B fragment layout: lanes 0-15 -> K={0,1}, lanes 16-31 -> K={2,3};
// row/col = lane & 15.  C/D: vgpr r -> row r (lanes 0-15) / r+8 (lanes 16-31).
// ---------------------------------------------------------------------------
#define CB     512
#define CWAVES (CB / 32)

__global__ __launch_bounds__(CB)
void uwe_sim_lse(const float* __restrict__ topic, const float* __restrict__ wemb,
                 const int* __restrict__ idx, const unsigned char* __restrict__ valid,
                 const int* __restrict__ flags, float* __restrict__ lossOut) {
    const int t    = blockIdx.x;
    const int tid  = threadIdx.x;
    const int wave = tid >> 5;
    const int lane = tid & 31;
    const int col16 = lane & 15;
    const int koff  = (lane >> 4) * 2;          // K offset within a 4-step

    __shared__ __align__(16) float sA[K_DIM * D_DIM];   // 40 KB topic_emb[t]
    __shared__ int           sIdx[M_DIM];
    __shared__ unsigned char sValid[M_DIM];
    __shared__ float sMx[CWAVES * 64];
    __shared__ float sSm[CWAVES * 64];
    __shared__ float sLse[64];

    const float* tbase = topic + (size_t)t * K_DIM * D_DIM;
    for (int i = tid; i < K_DIM * D_DIM; i += CB) sA[i] = tbase[i];
    for (int m = tid; m < M_DIM; m += CB) {
        sIdx[m]   = idx[t * M_DIM + m];
        sValid[m] = valid[t * M_DIM + m];
    }
    __syncthreads();

    // per-lane online logsumexp state: rows mt*16 + (laneHi?8:0) + r, own columns
    float mx[4][8], sm[4][8];
#pragma unroll
    for (int mt = 0; mt < 4; ++mt)
#pragma unroll
        for (int r = 0; r < 8; ++r) { mx[mt][r] = NEG_BIG; sm[mt][r] = 0.0f; }

    const v8f vzero = {};
    for (int wt = wave; wt < N_TILES; wt += CWAVES) {
        const int  n       = wt * 16 + col16;
        const bool inRange = (n < M_DIM);
        const int  w       = inRange ? sIdx[n] : 0;
        const bool colOk   = inRange && (sValid[n] != 0);
        const float* wptr  = wemb + (size_t)w * D_DIM + koff;

        v8f acc[4];
#pragma unroll
        for (int mt = 0; mt < 4; ++mt) acc[mt] = vzero;

        for (int d = 0; d < D_DIM / 4; ++d) {
            float2 bf2 = *(const float2*)(wptr + d * 4);
            v2f bfrag; bfrag[0] = bf2.x; bfrag[1] = bf2.y;
#pragma unroll
            for (int mt = 0; mt < 4; ++mt) {
                int arow = mt * 16 + col16;
                if (arow > K_DIM - 1) arow = K_DIM - 1;   // pad rows, discarded later
                float2 af2 = *(const float2*)&sA[arow * D_DIM + d * 4 + koff];
                v2f afrag; afrag[0] = af2.x; afrag[1] = af2.y;
                acc[mt] = __builtin_amdgcn_wmma_f32_16x16x4_f32(
                    false, afrag, false, bfrag, (short)0, acc[mt], false, false);
            }
        }

        // mask (invalid column -> -1e30, matching the reference) + online LSE
#pragma unroll
        for (int mt = 0; mt < 4; ++mt)
#pragma unroll
            for (int r = 0; r < 8; ++r) {
                float v  = colOk ? acc[mt][r] * INV_TEMP : NEG_BIG;
                float nm = fmaxf(mx[mt][r], v);
                sm[mt][r] = sm[mt][r] * __expf(mx[mt][r] - nm) + __expf(v - nm);
                mx[mt][r] = nm;
            }
    }

    // merge the 16 columns within each half-group (wave32: groups of 16)
#pragma unroll
    for (int off = 8; off >= 1; off >>= 1) {
#pragma unroll
        for (int mt = 0; mt < 4; ++mt)
#pragma unroll
            for (int r = 0; r < 8; ++r) {
                float omx = __shfl_xor(mx[mt][r], off, 16);
                float osm = __shfl_xor(sm[mt][r], off, 16);
                float nm  = fmaxf(mx[mt][r], omx);
                sm[mt][r] = sm[mt][r] * __expf(mx[mt][r] - nm) + osm * __expf(omx - nm);
                mx[mt][r] = nm;
            }
    }
    if (col16 == 0) {
        const int half = (lane >> 4) * 8;
#pragma unroll
        for (int mt = 0; mt < 4; ++mt)
#pragma unroll
            for (int r = 0; r < 8; ++r) {
                sMx[wave * 64 + mt * 16 + half + r] = mx[mt][r];
                sSm[wave * 64 + mt * 16 + half + r] = sm[mt][r];
            }
    }
    __syncthreads();

    if (tid < K_DIM) {
        float gm = NEG_BIG, gs = 0.0f;
        for (int wv = 0; wv < CWAVES; ++wv) {
            float omx = sMx[wv * 64 + tid], osm = sSm[wv * 64 + tid];
            float nm  = fmaxf(gm, omx);
            gs = gs * __expf(gm - nm) + osm * __expf(omx - nm);
            gm = nm;
        }
        sLse[tid] = gm + __logf(gs);
    }
    __syncthreads();
    if (tid == 0) {
        float s = 0.0f;
        for (int k = 0; k < K_DIM; ++k) s += sLse[k];
        lossOut[t] = flags[t] ? (s / (float)K_DIM) : 0.0f;
    }
}

// ---------------------------------------------------------------------------
// Kernel D: final scalar = sum(loss_t over valid t) / max(count,1)
// ---------------------------------------------------------------------------
__global__ void uwe_final(const float* __restrict__ loss, const int* __restrict__ flags,
                          float* __restrict__ out) {
    float s = 0.0f, c = 0.0f;
    if (threadIdx.x < T_DIM) {
        if (flags[threadIdx.x]) { s = loss[threadIdx.x]; c = 1.0f; }
    }
#pragma unroll
    for (int off = 16; off >= 1; off >>= 1) {
        s += __shfl_xor(s, off, 32);
        c += __shfl_xor(c, off, 32);
    }
    if (threadIdx.x == 0) out[0] = (c > 0.0f) ? s * (1.0f / fmaxf(c, 1.0f)) : 0.0f;
}

// ---------------------------------------------------------------------------
extern "C" void kernel_launch(void* const* d_in, const int* in_sizes, int n_in,
                              void* d_out, int out_size, void* d_ws, size_t ws_size,
                              hipStream_t stream) {
    const int*   wc    = (const int*)d_in[0];     // [T,V] int32
    const float* beta  = (const float*)d_in[1];   // [T,K,V] f32
    const float* topic = (const float*)d_in[2];   // [T,K,D] f32
    const float* wemb  = (const float*)d_in[3];   // [V,D] f32

    char* ws = (char*)d_ws;
    int*           wsIdx   = (int*)ws;                                           // 80000 B
    unsigned char* wsValid = (unsigned char*)(ws + T_DIM * M_DIM * sizeof(int)); // 20000 B
    float*         wsLoss  = (float*)(ws + T_DIM * M_DIM * (sizeof(int) + 1));   // @100000 (4-aligned)
    int*           wsFlags = (int*)(wsLoss + T_DIM);

    uwe_topk   <<<T_DIM * K_DIM, 256, 0, stream>>>(beta, wsIdx);
    uwe_valid  <<<T_DIM,         256, 0, stream>>>(wc, wsIdx, wsValid, wsFlags);
    uwe_sim_lse<<<T_DIM,         CB,  0, stream>>>(topic, wemb, wsIdx, wsValid, wsFlags, wsLoss);
    uwe_final  <<<1,             32,  0, stream>>>(wsLoss, wsFlags, (float*)d_out);
}